// SmartAdaptiveVQVAE_20315195310707
// MI455X (gfx1250) — compile-verified
//
#include <hip/hip_runtime.h>
#include <hip/hip_bf16.h>

typedef __attribute__((ext_vector_type(2))) float v2f;
typedef __attribute__((ext_vector_type(4))) float v4f;
typedef __attribute__((ext_vector_type(8))) float v8f;

#define K_CODES     512
#define DDIM        64
#define NROWS       (16 * 8192)          // 131072 rows
#define LDS_STRIDE  68                    // 64 + 4 pad: lanes over consecutive codes hit distinct banks
#define WAVES_PER_BLOCK 8
#define TILES_PER_WAVE  4
#define NUM_BLOCKS  ((NROWS / 16) / (WAVES_PER_BLOCK * TILES_PER_WAVE))  // 256

// d_out layout (reference tuple, flattened in return order):
//   [0]                    vq_loss
//   [1 .. 8388608]         quantized_st  (N*D = 8388608)
//   [8388609]              perplexity
//   [8388610 .. +N*K-1]    encodings     (N*K = 67108864)
//   [75497474 .. +N-1]     encoding_indices (as float)
#define OFF_Q   1
#define OFF_PPL 8388609
#define OFF_ENC 8388610
#define OFF_IDX 75497474

// workspace: ws[0] = sum((q-x)^2), ws[1..512] = per-code counts (float)
__global__ void vq_init(float* __restrict__ ws) {
    for (int i = threadIdx.x; i < 1 + K_CODES; i += blockDim.x) ws[i] = 0.0f;
}

__global__ __launch_bounds__(256, 1)
void vq_main(const float* __restrict__ x, const float* __restrict__ cb,
             float* __restrict__ out, float* __restrict__ ws) {
    extern __shared__ float lds[];
    float* scb   = lds;                          // 512 * 68 floats
    float* snorm = lds + K_CODES * LDS_STRIDE;   // 512 floats

    const int tid = threadIdx.x;

    // Stage codebook into LDS (padded rows), coalesced global reads.
    for (int i = tid; i < K_CODES * DDIM; i += blockDim.x) {
        int code = i >> 6, d = i & 63;
        scb[code * LDS_STRIDE + d] = cb[i];
    }
    __syncthreads();
    // Per-code squared norms.
    for (int code = tid; code < K_CODES; code += blockDim.x) {
        float s = 0.0f;
        const float* r = scb + code * LDS_STRIDE;
        #pragma unroll
        for (int d = 0; d < DDIM; ++d) { float v = r[d]; s += v * v; }
        snorm[code] = s;
    }
    __syncthreads();

    const int lane = tid & 31;
    const int wave = tid >> 5;
    const int m    = lane & 15;     // row (A) / code (B) index within tile
    const int hi   = lane >> 4;     // half-wave select
    const int koff = hi * 2;        // K sub-offset for 16x16x4 f32 operand layout

    float* outQ = out + OFF_Q;
    float* outE = out + OFF_ENC;
    float* outI = out + OFF_IDX;

    float lsum = 0.0f;

    for (int t = 0; t < TILES_PER_WAVE; ++t) {
        const int tile = (blockIdx.x * WAVES_PER_BLOCK + wave) * TILES_PER_WAVE + t;
        const int row0 = tile * 16;

        // Prefetch next tile's input rows into cache (global_prefetch_b8).
        if (t + 1 < TILES_PER_WAVE) {
            const float* nf = x + (size_t)(row0 + 16) * DDIM + lane * 32;
            __builtin_prefetch(nf, 0, 1);
        }

        // A fragments: 16 x (16x4) f32 chunks covering D=64.
        // Lane layout per ISA: lanes 0-15 hold {K=4kc+0, 4kc+1}, lanes 16-31 {K=4kc+2, 4kc+3}, M = lane&15.
        v2f a[16];
        #pragma unroll
        for (int kc = 0; kc < 16; ++kc) {
            const float* p = x + (size_t)(row0 + m) * DDIM + kc * 4 + koff;
            a[kc].x = p[0];
            a[kc].y = p[1];
        }

        float best[8];
        int   bidx[8];
        #pragma unroll
        for (int r = 0; r < 8; ++r) { best[r] = 3.4e38f; bidx[r] = 0; }

        // Sweep 32 code tiles of 16 codes, two at a time with independent
        // accumulator chains so the two WMMA streams interleave and hide
        // each other's LDS latency (only 2 waves/SIMD at this LDS footprint).
        for (int n0 = 0; n0 < K_CODES; n0 += 32) {
            const int na = n0 + m;                       // code column, tile A
            const int nb = n0 + 16 + m;                  // code column, tile B
            const float* browA = scb + na * LDS_STRIDE;  // padded -> conflict-free ds_load_2addr
            const float* browB = scb + nb * LDS_STRIDE;

            v8f cA = {0.f, 0.f, 0.f, 0.f, 0.f, 0.f, 0.f, 0.f};
            v8f cB = {0.f, 0.f, 0.f, 0.f, 0.f, 0.f, 0.f, 0.f};
            #pragma unroll
            for (int kc = 0; kc < 16; ++kc) {
                v2f ba, bb;
                ba.x = browA[kc * 4 + koff];
                ba.y = browA[kc * 4 + koff + 1];
                bb.x = browB[kc * 4 + koff];
                bb.y = browB[kc * 4 + koff + 1];
                cA = __builtin_amdgcn_wmma_f32_16x16x4_f32(
                         false, a[kc], false, ba, (short)0, cA, false, false);
                cB = __builtin_amdgcn_wmma_f32_16x16x4_f32(
                         false, a[kc], false, bb, (short)0, cB, false, false);
            }

            const float cnA = snorm[na];                 // same code for all 8 C rows
            const float cnB = snorm[nb];
            #pragma unroll
            for (int r = 0; r < 8; ++r) {
                float sA = cnA - 2.0f * cA[r];           // ||e||^2 - 2 x.e  (||x||^2 constant per row)
                float sB = cnB - 2.0f * cB[r];
                if (sA < best[r]) { best[r] = sA; bidx[r] = na; }
                if (sB < best[r]) { best[r] = sB; bidx[r] = nb; }
            }
        }

        // Argmin butterfly across the 16-lane groups (tie-break: lowest index, matching jnp.argmin).
        #pragma unroll
        for (int off = 1; off < 16; off <<= 1) {
            #pragma unroll
            for (int r = 0; r < 8; ++r) {
                float ob = __shfl_xor(best[r], off, 16);
                int   oi = __shfl_xor(bidx[r], off, 16);
                if (ob < best[r] || (ob == best[r] && oi < bidx[r])) {
                    best[r] = ob; bidx[r] = oi;
                }
            }
        }

        // Outputs: group hi handles rows row0 + hi*8 + r, 16 lanes cooperate per row.
        #pragma unroll
        for (int r = 0; r < 8; ++r) {
            const int row = row0 + hi * 8 + r;
            const int idx = bidx[r];

            // quantized_st = codebook[idx] (numerically); fused loss accumulation.
            const float* q  = scb + idx * LDS_STRIDE + 4 * m;
            const float* xi = x + (size_t)row * DDIM + 4 * m;
            v4f qv; qv.x = q[0]; qv.y = q[1]; qv.z = q[2]; qv.w = q[3];
            float d0 = qv.x - xi[0], d1 = qv.y - xi[1];
            float d2 = qv.z - xi[2], d3 = qv.w - xi[3];
            lsum += d0 * d0 + d1 * d1 + d2 * d2 + d3 * d3;
            *(v4f*)(outQ + (size_t)row * DDIM + 4 * m) = qv;

            // one-hot encodings row: 8 coalesced float4 stores per lane group.
            #pragma unroll
            for (int j = 0; j < 8; ++j) {
                const int c0 = j * 64 + 4 * m;
                v4f e;
                e.x = (c0     == idx) ? 1.0f : 0.0f;
                e.y = (c0 + 1 == idx) ? 1.0f : 0.0f;
                e.z = (c0 + 2 == idx) ? 1.0f : 0.0f;
                e.w = (c0 + 3 == idx) ? 1.0f : 0.0f;
                *(v4f*)(outE + (size_t)row * K_CODES + c0) = e;
            }

            if (m == 0) {
                outI[row] = (float)idx;
                atomicAdd(ws + 1 + idx, 1.0f);   // histogram for perplexity
            }
        }
    }

    atomicAdd(ws, lsum);
}

__global__ void vq_final(const float* __restrict__ ws, float* __restrict__ out) {
    __shared__ float red[K_CODES];
    const int t = threadIdx.x;
    float p = ws[1 + t] / (float)NROWS;          // avg_probs[k]
    red[t] = p * logf(p + 1e-10f);
    __syncthreads();
    for (int s = K_CODES / 2; s > 0; s >>= 1) {
        if (t < s) red[t] += red[t + s];
        __syncthreads();
    }
    if (t == 0) {
        out[OFF_PPL] = expf(-red[0]);
        // q_latent_loss == e_latent_loss numerically -> (1 + 0.25) * mean((q - x)^2)
        out[0] = 1.25f * ws[0] / (float)(NROWS * DDIM);
    }
}

extern "C" void kernel_launch(void* const* d_in, const int* in_sizes, int n_in,
                              void* d_out, int out_size, void* d_ws, size_t ws_size,
                              hipStream_t stream) {
    const float* x  = (const float*)d_in[0];   // [16, 8192, 64]
    const float* cb = (const float*)d_in[1];   // [512, 64]
    float* out = (float*)d_out;
    float* ws  = (float*)d_ws;

    hipLaunchKernelGGL(vq_init, dim3(1), dim3(256), 0, stream, ws);

    const size_t shbytes = (size_t)(K_CODES * LDS_STRIDE + K_CODES) * sizeof(float); // ~138 KB of 320 KB WGP LDS
    hipLaunchKernelGGL(vq_main, dim3(NUM_BLOCKS), dim3(256), shbytes, stream, x, cb, out, ws);

    hipLaunchKernelGGL(vq_final, dim3(1), dim3(K_CODES), 0, stream, ws, out);
}